// Pixelwise_15530601742395
// MI455X (gfx1250) — compile-verified
//
#include <hip/hip_runtime.h>
#include <math.h>

#define N_CORRS 10000
#define NPIX    16384

static constexpr double D_SPEED   = 299792458.0 * 1000.0;
static constexpr double D_DMAX    = 10000.0;
static constexpr double D_FMAX    = D_SPEED / (2.0 * D_DMAX);
static constexpr double D_TAUMIN  = 1.0 / D_FMAX;
static constexpr double D_DT      = D_TAUMIN / (double)N_CORRS;
static constexpr double D_PSRC    = 1.0e9;
static constexpr double D_PAMB    = 1.0e6;
static constexpr double D_BFACT   = 1.0 / (D_PAMB + D_PSRC);   // GAMMA*MEAN_BETA*T_INT

typedef __attribute__((ext_vector_type(2))) float v2f;
typedef __attribute__((ext_vector_type(8))) float v8f;

// ---------------------------------------------------------------- col sums
__global__ __launch_bounds__(256)
void colsum_kernel(const float* __restrict__ Mod, const float* __restrict__ Demod,
                   float* __restrict__ stats) {
    const int k = blockIdx.x;            // 0..2
    if (k == 0 && threadIdx.x == 0) stats[15] = 0.f;   // reset row-norm max slot
    __shared__ float red[256];
    float sm = 0.f, sd = 0.f;
    for (int n = threadIdx.x; n < N_CORRS; n += 256) {
        sm += Mod[n * 3 + k];
        sd += Demod[n * 3 + k];
    }
    red[threadIdx.x] = sm; __syncthreads();
    for (int o = 128; o > 0; o >>= 1) { if (threadIdx.x < o) red[threadIdx.x] += red[threadIdx.x + o]; __syncthreads(); }
    if (threadIdx.x == 0) stats[k] = red[0];
    __syncthreads();
    red[threadIdx.x] = sd; __syncthreads();
    for (int o = 128; o > 0; o >>= 1) { if (threadIdx.x < o) red[threadIdx.x] += red[threadIdx.x + o]; __syncthreads(); }
    if (threadIdx.x == 0) stats[4 + k] = red[0];
}

// ------------------------------------------------- direct circular correlation
// CorrRawPad[n*4+k] = DT * sum_m Mod[m,k] * Demod[(m+n) mod N, k]
__global__ __launch_bounds__(256)
void corr_kernel(const float* __restrict__ Mod, const float* __restrict__ Demod,
                 float* __restrict__ CorrRawPad) {
    const int n = blockIdx.x;
    __shared__ float red[256];
    float acc0 = 0.f, acc1 = 0.f, acc2 = 0.f;
    for (int m = threadIdx.x; m < N_CORRS; m += 256) {
        int j = m + n; if (j >= N_CORRS) j -= N_CORRS;
        acc0 += Mod[m * 3 + 0] * Demod[j * 3 + 0];
        acc1 += Mod[m * 3 + 1] * Demod[j * 3 + 1];
        acc2 += Mod[m * 3 + 2] * Demod[j * 3 + 2];
    }
    float acc[3] = {acc0, acc1, acc2};
    #pragma unroll
    for (int k = 0; k < 3; ++k) {
        red[threadIdx.x] = acc[k]; __syncthreads();
        for (int o = 128; o > 0; o >>= 1) { if (threadIdx.x < o) red[threadIdx.x] += red[threadIdx.x + o]; __syncthreads(); }
        if (threadIdx.x == 0) CorrRawPad[n * 4 + k] = red[0] * (float)D_DT;
        __syncthreads();
    }
    if (threadIdx.x == 0) CorrRawPad[n * 4 + 3] = 0.f;
}

// ------------------------------------------ per-column mean / unbiased std
__global__ __launch_bounds__(256)
void meanstd_kernel(const float* __restrict__ CorrRawPad, float* __restrict__ stats) {
    const int k = blockIdx.x;
    __shared__ float red[256];
    float acc = 0.f;
    for (int n = threadIdx.x; n < N_CORRS; n += 256) acc += CorrRawPad[n * 4 + k];
    red[threadIdx.x] = acc; __syncthreads();
    for (int o = 128; o > 0; o >>= 1) { if (threadIdx.x < o) red[threadIdx.x] += red[threadIdx.x + o]; __syncthreads(); }
    const float mean = red[0] / (float)N_CORRS;
    __syncthreads();
    acc = 0.f;
    for (int n = threadIdx.x; n < N_CORRS; n += 256) {
        float d = CorrRawPad[n * 4 + k] - mean;
        acc += d * d;
    }
    red[threadIdx.x] = acc; __syncthreads();
    for (int o = 128; o > 0; o >>= 1) { if (threadIdx.x < o) red[threadIdx.x] += red[threadIdx.x + o]; __syncthreads(); }
    if (threadIdx.x == 0) {
        float sd  = sqrtf(red[0] / (float)(N_CORRS - 1));   // ddof=1
        float sgn = (stats[k] >= 0.f) ? 1.f : -1.f;         // sign of scale s_k
        stats[8 + k]  = mean;
        stats[12 + k] = sgn / sd;
    }
}

// ----------------------- write NormCorrPad + global max row-norm^2 (for shift)
__global__ __launch_bounds__(256)
void normcorr_kernel(const float* __restrict__ CorrRawPad, float* __restrict__ stats,
                     float* __restrict__ NormCorrPad) {
    const int n = blockIdx.x * blockDim.x + threadIdx.x;
    if (n >= N_CORRS) return;
    float rn = 0.f;
    #pragma unroll
    for (int k = 0; k < 3; ++k) {
        float v = (CorrRawPad[n * 4 + k] - stats[8 + k]) * stats[12 + k];
        NormCorrPad[n * 4 + k] = v;
        rn = fmaf(v, v, rn);
    }
    NormCorrPad[n * 4 + 3] = 0.f;
    // nonnegative floats compare correctly as ints; atomic max is order-independent
    atomicMax(reinterpret_cast<int*>(stats) + 15, __float_as_int(rn));
}

// -------------------------------------------- per-pixel gather + K-normalize
__global__ __launch_bounds__(256)
void pixprep_kernel(const float* __restrict__ gt, const float* __restrict__ CorrRawPad,
                    const float* __restrict__ stats, float* __restrict__ NormBPad) {
    const int p = blockIdx.x * blockDim.x + threadIdx.x;
    if (p >= NPIX) return;
    int idx = (int)rintf(gt[p]);                       // round-half-even like jnp.round
    idx = min(max(idx, 0), N_CORRS - 1);
    const float f = (float)D_BFACT;
    float B[3];
    #pragma unroll
    for (int k = 0; k < 3; ++k) {
        float kappa = stats[4 + k] * (float)D_DT;      // sum(Demod_k)*DT
        B[k] = f * (CorrRawPad[idx * 4 + k] + (float)D_PAMB * kappa);
    }
    float mu = (B[0] + B[1] + B[2]) * (1.f / 3.f);
    float v0 = B[0] - mu, v1 = B[1] - mu, v2 = B[2] - mu;
    float sd = sqrtf((v0 * v0 + v1 * v1 + v2 * v2) * 0.5f);   // ddof=1, K=3
    float inv = 1.f / sd;
    NormBPad[p * 4 + 0] = v0 * inv;
    NormBPad[p * 4 + 1] = v1 * inv;
    NormBPad[p * 4 + 2] = v2 * inv;
    NormBPad[p * 4 + 3] = 0.f;
}

// ------------------- fixed-shift chunk accumulator (3 VALU + 1 TRANS per elem)
__device__ __forceinline__ void chunk_accum(const v8f& c, float nb, float log2e,
                                            float ebase, float& s, float& t) {
    float e[8];
    #pragma unroll
    for (int r = 0; r < 8; ++r)
        e[r] = __builtin_amdgcn_exp2f(fmaf(c[r], log2e, ebase));   // exp(z - SHIFT)
    float cs = ((e[0] + e[1]) + (e[2] + e[3])) + ((e[4] + e[5]) + (e[6] + e[7]));
    float cu = e[1];
    #pragma unroll
    for (int r = 2; r < 8; ++r) cu = fmaf(e[r], (float)r, cu);     // sum e_r * r
    s += cs;
    t = fmaf(cs, nb, t) + cu;                                      // sum e_r * (nb + r)
}

// ---------------- fused WMMA zncc + fixed-shift softmax soft-argmax (wave32)
// Per wave: 16 pixels. A = NormCorr tile (16 samples x 4), B = NormB^T (4 x 16 px).
// D layout: lane L -> pixel col (L&15); VGPR r -> sample row r + 8*(L>=16).
__global__ __launch_bounds__(256)
void zncc_softargmax_kernel(const float* __restrict__ NormCorrPad,
                            const float* __restrict__ NormBPad,
                            const float* __restrict__ stats,
                            float* __restrict__ out) {
    const int lane    = threadIdx.x & 31;
    const int waveId  = (blockIdx.x * blockDim.x + threadIdx.x) >> 5;
    const int pixBase = waveId * 16;
    const int col     = lane & 15;
    const int khalf   = (lane >> 4) << 1;      // K pair: {0,1} or {2,3}
    const int p       = pixBase + col;

    // |z| <= ||NormB||_2 * max_n ||NormCorr_n||_2 = sqrt(2 * max row-norm^2)
    const float log2e = 1.4426950408889634f;
    const float shift = sqrtf(2.0f * stats[15]);
    const float ebase = -shift * log2e;

    // B fragment: constant over entire sample loop (heavy reuse)
    v2f b;
    b.x = NormBPad[p * 4 + khalf];
    b.y = NormBPad[p * 4 + khalf + 1];

    const float rowOff = (float)((lane >> 4) * 8);
    float s = 0.f, t = 0.f;

    // software-pipelined A-tile stream
    const float* ap0 = &NormCorrPad[col * 4 + khalf];
    v2f a; a.x = ap0[0]; a.y = ap0[1];

    for (int n0 = 0; n0 < N_CORRS - 16; n0 += 16) {
        v2f acur = a;
        const float* apn = &NormCorrPad[(n0 + 16 + col) * 4 + khalf];
        a.x = apn[0]; a.y = apn[1];
        __builtin_prefetch(&NormCorrPad[(n0 + 32 + col) * 4], 0, 1);

        v8f c = {};
        c = __builtin_amdgcn_wmma_f32_16x16x4_f32(
                false, acur, false, b, (short)0, c, false, false);
        chunk_accum(c, (float)n0 + rowOff, log2e, ebase, s, t);
    }
    {   // epilogue chunk
        v8f c = {};
        c = __builtin_amdgcn_wmma_f32_16x16x4_f32(
                false, a, false, b, (short)0, c, false, false);
        chunk_accum(c, (float)(N_CORRS - 16) + rowOff, log2e, ebase, s, t);
    }

    // merge the two lanes holding the same pixel (lane ^ 16)
    float S = s + __shfl_xor(s, 16);
    float T = t + __shfl_xor(t, 16);
    if (lane < 16) out[p] = T / S;
}

// ----------------------------------------------------------------- launcher
extern "C" void kernel_launch(void* const* d_in, const int* in_sizes, int n_in,
                              void* d_out, int out_size, void* d_ws, size_t ws_size,
                              hipStream_t stream) {
    const float* gt    = (const float*)d_in[0];   // (1,128,128)
    const float* Mod   = (const float*)d_in[1];   // (10000,3)
    const float* Demod = (const float*)d_in[2];   // (10000,3)
    float* ws = (float*)d_ws;

    float* CorrRawPad  = ws;                      // 40000 floats
    float* NormCorrPad = ws + 40000;              // 40000 floats
    float* NormBPad    = ws + 80000;              // 65536 floats
    float* stats       = ws + 145536;             // 16 floats
    float* out         = (float*)d_out;           // 16384 floats

    hipLaunchKernelGGL(colsum_kernel,   dim3(3),                 dim3(256), 0, stream, Mod, Demod, stats);
    hipLaunchKernelGGL(corr_kernel,     dim3(N_CORRS),           dim3(256), 0, stream, Mod, Demod, CorrRawPad);
    hipLaunchKernelGGL(meanstd_kernel,  dim3(3),                 dim3(256), 0, stream, CorrRawPad, stats);
    hipLaunchKernelGGL(normcorr_kernel, dim3((N_CORRS + 255) / 256), dim3(256), 0, stream, CorrRawPad, stats, NormCorrPad);
    hipLaunchKernelGGL(pixprep_kernel,  dim3(NPIX / 256),        dim3(256), 0, stream, gt, CorrRawPad, stats, NormBPad);
    hipLaunchKernelGGL(zncc_softargmax_kernel, dim3(NPIX / 128), dim3(256), 0, stream, NormCorrPad, NormBPad, stats, out);
}